// MyModel_47029891891899
// MI455X (gfx1250) — compile-verified
//
#include <hip/hip_runtime.h>

// group_points gather: out[b,c,p,s] = feature[b, c, idx[b,p,s]]
// B=4, C=256, N=16384, NPOINT=4096, NSAMPLE=32
#define B_       4
#define C_       256
#define N_       16384
#define NPOINT_  4096
#define NSAMPLE_ 32

typedef float __attribute__((ext_vector_type(4))) f4;
typedef int   __attribute__((ext_vector_type(4))) i4;

__global__ __launch_bounds__(256) void group_points_gather_kernel(
    const float* __restrict__ feature,
    const int*   __restrict__ idx,
    float*       __restrict__ out)
{
    __shared__ int sidx[NSAMPLE_];

    const int bp   = blockIdx.x;          // one block per (b, p)
    const int b    = bp >> 12;            // / NPOINT (4096 = 2^12)
    const int p    = bp & (NPOINT_ - 1);  // % NPOINT
    const int t    = threadIdx.x;
    const int wave = t >> 5;              // 8 waves (wave32)
    const int lane = t & 31;

    // ---- Stage the 32 gather indices into LDS via the CDNA5 async copy path.
    // Wave 0 (all 32 lanes active -> EXEC all-ones) issues one async b32 per lane.
    if (t < NSAMPLE_) {
        const int* g = idx + ((size_t)b * NPOINT_ + p) * NSAMPLE_ + t;
#if __has_builtin(__builtin_amdgcn_global_load_async_to_lds_b32)
        __builtin_amdgcn_global_load_async_to_lds_b32(
            (__attribute__((address_space(1))) int*)g,
            (__attribute__((address_space(3))) int*)&sidx[t],
            /*offset=*/0, /*cpol=*/0);
#else
        unsigned lds_addr = (unsigned)(size_t)&sidx[t];   // low 32 bits of flat = LDS offset
        unsigned long long gaddr = (unsigned long long)(size_t)g;
        asm volatile("global_load_async_to_lds_b32 %0, %1, off"
                     :: "v"(lds_addr), "v"(gaddr) : "memory");
#endif
    }
    // Issuing wave drains ASYNCcnt; other waves have ASYNCcnt==0 so this is free.
#if __has_builtin(__builtin_amdgcn_s_wait_asynccnt)
    __builtin_amdgcn_s_wait_asynccnt(0);
#else
    asm volatile("s_wait_asynccnt 0x0" ::: "memory");
#endif
    __syncthreads();

    // ---- Lane layout: lane handles 4 consecutive samples and one channel row.
    // A wave iteration covers 4 channels x 32 samples; 8 iters x 8 waves = 256 channels.
    const int s0   = (lane & 7) * 4;  // sample group: 0,4,...,28
    const int crow = lane >> 3;       // channel row within wave: 0..3

    const i4 nn = *(const i4*)&sidx[s0];   // ds_load_b128 (16B aligned)

    const size_t fbase = (size_t)b * C_ * N_;

#pragma unroll
    for (int it = 0; it < 8; ++it) {
        const int c = wave * 4 + it * 32 + crow;
        const float* __restrict__ frow = feature + fbase + (size_t)c * N_;

        // Four scattered b32 gathers within a 64KB row -> L2 hits (feature fits in 192MB L2).
        f4 v;
        v.x = frow[nn.x];
        v.y = frow[nn.y];
        v.z = frow[nn.z];
        v.w = frow[nn.w];

        // Coalesced 16B non-temporal store: output is streamed once, keep it out of L2.
        float* o = out + ((size_t)(b * C_ + c) * NPOINT_ + p) * NSAMPLE_ + s0;
        __builtin_nontemporal_store(v, (f4*)o);
    }
}

extern "C" void kernel_launch(void* const* d_in, const int* in_sizes, int n_in,
                              void* d_out, int out_size, void* d_ws, size_t ws_size,
                              hipStream_t stream) {
    const float* feature = (const float*)d_in[0];
    const int*   idx     = (const int*)d_in[1];
    float*       out     = (float*)d_out;

    dim3 grid(B_ * NPOINT_);   // 16384 blocks: one per (b, p)
    dim3 block(256);           // 8 wave32 waves
    hipLaunchKernelGGL(group_points_gather_kernel, grid, block, 0, stream,
                       feature, idx, out);
}